// SingleHashNMSKPtC_16338055594732
// MI455X (gfx1250) — compile-verified
//
#include <hip/hip_runtime.h>
#include <stdint.h>

// ---------------------------------------------------------------------------
// Hash-NMS for MI455X (gfx1250).
// Memory/atomic-bound: hash table (<=128MB) is L2-resident (192MB L2);
// streams use NT hints + async LDS data mover (ASYNCcnt path) so they do
// not evict the table. No contraction in this workload -> no WMMA by design.
// ---------------------------------------------------------------------------

typedef float vfloat4 __attribute__((ext_vector_type(4)));

// log(w/16)/log(0.7) == log2(w/16) * (1/log2(0.7))
#define HNMS_INV_LOG2_ALPHA (-1.94341641133349073f) /* 1/log2(0.7) */
#define HNMS_LOG2_ALPHA     (-0.51457317282975824f) /* log2(0.7)   */
#define HNMS_CELL0 (9.6f)                           /* GAMMA * W0 = 0.6*16 */
#define HNMS_EMPTY 0xFFFFFFFFFFFFFFFFull

__device__ __forceinline__ uint32_t hnms_pack(int a, int b) {
    a += 4096; b += 4096;                      // CODE_OFF
    a = a < 0 ? 0 : (a > 8191 ? 8191 : a);     // clip to [0, CODE_RANGE)
    b = b < 0 ? 0 : (b > 8191 ? 8191 : b);
    return ((uint32_t)a << 13) | (uint32_t)b;  // a * 8192 + b, < 2^26
}

__device__ __forceinline__ uint64_t hnms_key(vfloat4 r) {
    float cx = r.x, cy = r.y, w = r.z, h = r.w;
    // round-half-even == v_rndne_f32; v_log_f32 computes log2 natively
    float fiw = __builtin_rintf(__builtin_amdgcn_logf(w * 0.0625f) *
                                HNMS_INV_LOG2_ALPHA);
    float fih = __builtin_rintf(__builtin_amdgcn_logf(h * 0.0625f) *
                                HNMS_INV_LOG2_ALPHA);
    // alpha^iw = exp2(iw * log2(alpha)); v_exp_f32 computes exp2 natively
    float cw = HNMS_CELL0 * __builtin_amdgcn_exp2f(fiw * HNMS_LOG2_ALPHA);
    float ch = HNMS_CELL0 * __builtin_amdgcn_exp2f(fih * HNMS_LOG2_ALPHA);
    int ix = (int)__builtin_rintf(cx / cw - 0.5f);   // keep true f32 divide
    int iy = (int)__builtin_rintf(cy / ch - 0.5f);
    uint32_t hi = hnms_pack(ix, iy);
    uint32_t lo = hnms_pack((int)fiw, (int)fih);
    return ((uint64_t)hi << 26) | (uint64_t)lo;      // 52-bit bucket id
}

// 32-bit mixer over the two 26-bit halves: single-cycle v_mul_lo_u32 only
// (slot index needs ~23 bits; full 52-bit key is still stored/compared).
__device__ __forceinline__ uint32_t hnms_hash32(uint64_t key) {
    uint32_t h = (uint32_t)key * 0x85EBCA77u
               ^ (uint32_t)(key >> 26) * 0x9E3779B1u;
    h ^= h >> 16; h *= 0x7FEB352Du;
    h ^= h >> 15; h *= 0x846CA68Bu;
    h ^= h >> 16;
    return h;
}

// conf > 0 always (uniform in (0.01,1)) -> float bits are order-preserving.
// Tie on conf -> larger ~i wins == smaller index wins (matches lexsort).
__device__ __forceinline__ uint64_t hnms_val(float c, int i) {
    return ((uint64_t)__float_as_uint(c) << 32) | (uint32_t)(~(uint32_t)i);
}

__global__ void __launch_bounds__(256)
hnms_init(unsigned long long* __restrict__ keys,
          unsigned long long* __restrict__ vals, uint32_t cap) {
    uint32_t i = blockIdx.x * 256u + threadIdx.x;
    if (i < cap) {
        keys[i] = HNMS_EMPTY;
        vals[i] = 0ull;
    }
}

__global__ void __launch_bounds__(256)
hnms_insert(const float* __restrict__ rects, const float* __restrict__ conf,
            unsigned long long* __restrict__ keys,
            unsigned long long* __restrict__ vals, uint32_t mask, int n) {
    __shared__ __align__(16) vfloat4 tile[256];
    int i = blockIdx.x * 256 + (int)threadIdx.x;
    if (i >= n) return;

    // Stage this lane's 16B rect through LDS with the async data mover
    // (gfx1250 ASYNCcnt path); generic LDS pointer low 32 bits == LDS offset.
    unsigned lds_addr = (unsigned)(size_t)(&tile[threadIdx.x]);
    unsigned voff     = (unsigned)i * 16u;             // byte offset, fits u32
    uint64_t sbase    = (uint64_t)(uintptr_t)rects;    // SGPR-pair base (GVS)
    asm volatile("global_load_async_to_lds_b128 %0, %1, %2"
                 :: "v"(lds_addr), "v"(voff), "s"(sbase) : "memory");

    float c = __builtin_nontemporal_load(conf + i);    // NT: keep L2 for table

    asm volatile("s_wait_asynccnt 0x0" ::: "memory");  // own slot is ready
    vfloat4 r = tile[threadIdx.x];

    uint64_t key    = hnms_key(r);
    uint64_t packed = hnms_val(c, i);
    uint32_t slot   = hnms_hash32(key) & mask;

    // Open addressing: claim slot with CAS, then max-reduce the winner.
    // 64-bit atomics execute at the L2 atomic units (table is L2-resident).
    for (uint32_t p = 0; p <= mask; ++p) {
        unsigned long long prev =
            atomicCAS(&keys[slot], (unsigned long long)HNMS_EMPTY,
                      (unsigned long long)key);
        if (prev == HNMS_EMPTY || prev == (unsigned long long)key) {
            atomicMax(&vals[slot], (unsigned long long)packed);
            return;
        }
        slot = (slot + 1) & mask;
    }
}

__global__ void __launch_bounds__(256)
hnms_resolve(const float* __restrict__ rects, const float* __restrict__ conf,
             const unsigned long long* __restrict__ keys,
             const unsigned long long* __restrict__ vals, uint32_t mask, int n,
             float* __restrict__ out) {
    int i = blockIdx.x * 256 + (int)threadIdx.x;
    if (i >= n) return;

    const vfloat4* r4 = (const vfloat4*)rects;
    vfloat4 r = __builtin_nontemporal_load(r4 + i);    // NT b128 stream
    float   c = __builtin_nontemporal_load(conf + i);

    uint64_t key    = hnms_key(r);                     // deterministic recompute
    uint64_t packed = hnms_val(c, i);
    uint32_t slot   = hnms_hash32(key) & mask;

    float keepf = 0.0f;
    for (uint32_t p = 0; p <= mask; ++p) {
        unsigned long long k = keys[slot];
        if (k == (unsigned long long)key) {
            keepf = (vals[slot] == (unsigned long long)packed) ? 1.0f : 0.0f;
            break;
        }
        if (k == HNMS_EMPTY) break;                    // not found (shouldn't)
        slot = (slot + 1) & mask;
    }

    __builtin_nontemporal_store(keepf * c, out + i);       // kept_conf
    __builtin_nontemporal_store(keepf,     out + n + i);   // keep (as 0/1)
}

extern "C" void kernel_launch(void* const* d_in, const int* in_sizes, int n_in,
                              void* d_out, int out_size, void* d_ws,
                              size_t ws_size, hipStream_t stream) {
    const float* rects = (const float*)d_in[0];  // (N,4) row-major
    const float* conf  = (const float*)d_in[1];  // (N,)
    int n = in_sizes[1];
    float* out = (float*)d_out;                  // [kept_conf(N) | keep(N)]

    // Hash table: keys[cap] + vals[cap] (16 B/slot). Largest pow2 that fits
    // ws, capped at 8M slots (128 MB) — still inside the 192 MB L2.
    size_t cap = 1;
    while (((cap << 1) * 16ull) <= ws_size && cap < (1ull << 23)) cap <<= 1;
    if (cap < 2) cap = 2;
    unsigned long long* keys = (unsigned long long*)d_ws;
    unsigned long long* vals = keys + cap;
    uint32_t mask = (uint32_t)cap - 1u;

    const int tb = 256;  // 8 wave32 waves per workgroup
    hnms_init<<<(int)((cap + tb - 1) / tb), tb, 0, stream>>>(keys, vals,
                                                             (uint32_t)cap);
    int nb = (n + tb - 1) / tb;
    hnms_insert<<<nb, tb, 0, stream>>>(rects, conf, keys, vals, mask, n);
    hnms_resolve<<<nb, tb, 0, stream>>>(rects, conf, keys, vals, mask, n, out);
}